// MultiLossFactory_46188078301466
// MI455X (gfx1250) — compile-verified
//
#include <hip/hip_runtime.h>
#include <hip/hip_bf16.h>

// ---------------- problem constants ----------------
#define B_   16
#define C_   26
#define H_   256
#define W_   256
#define D_   (C_ * H_ * W_)       // 1,703,936 elems per batch (hm / bu)
#define D4_  (D_ / 4)             // 425,984 float4 per batch
#define DHW_ (2 * H_ * W_)        // 131,072 (hw_pred per batch)
#define P_   20
#define K_   2
#define E_   26
#define BPB_ 208                  // blocks per batch for kernel 1
#define ITERS_ (D4_ / (BPB_ * 256))   // = 8, exact
#define EPS_ 1e-6f

typedef __attribute__((ext_vector_type(2))) float v2f;
typedef __attribute__((ext_vector_type(8))) float v8f;

// -------------------------------------------------------------------
// Reduce 256 f32 values in LDS to a single sum using chained
// V_WMMA_F32_16X16X4_F32 with B = ones:  D[m][n] = C[m][n] + sum_k A[m][k].
// A (16x4 f32) layout: lane m (0..15) holds K=0 (vgpr a0), K=1 (a1);
// lane m+16 holds K=2, K=3.  Tile t covers s[64t + {2m,2m+1,2m+32,2m+33}],
// so 4 chained tiles cover all 256 indices exactly once.
// Must be called by a full wave (all 32 lanes active).
// -------------------------------------------------------------------
__device__ __forceinline__ float wmma_reduce256(const float* __restrict__ s, int lane) {
    v8f acc = {0.f, 0.f, 0.f, 0.f, 0.f, 0.f, 0.f, 0.f};
    v2f ones; ones[0] = 1.0f; ones[1] = 1.0f;
#pragma unroll
    for (int t = 0; t < 4; ++t) {
        v2f a;
        a[0] = s[t * 64 + 2 * lane];
        a[1] = s[t * 64 + 2 * lane + 1];
        acc = __builtin_amdgcn_wmma_f32_16x16x4_f32(
            /*neg_a=*/false, a, /*neg_b=*/false, ones,
            /*c_mod=*/(short)0, acc, /*reuse_a=*/false, /*reuse_b=*/false);
    }
    // D vgpr v: lanes 0-15 = row v, lanes 16-31 = row v+8 (columns identical).
    float r = acc[0] + acc[1] + acc[2] + acc[3] + acc[4] + acc[5] + acc[6] + acc[7];
    r += __shfl_xor(r, 16, 32);   // fold rows 0-7 half with rows 8-15 half
    return r;                     // total in all lanes
}

__device__ __forceinline__ float fast_sigmoid(float x) {
    float e = __expf(-x);
    return __builtin_amdgcn_rcpf(1.0f + e);
}

// -------------------------------------------------------------------
// Kernel 1: streaming focal-loss partial sums over hm (bandwidth-bound).
// grid = (BPB_, B_), block = 256.  Each thread: 8 iterations of float4.
// -------------------------------------------------------------------
__global__ __launch_bounds__(256)
void hm_focal_partial(const float* __restrict__ pred,
                      const float* __restrict__ gt,
                      float* __restrict__ partials) {
    const int b   = blockIdx.y;
    const int tid = threadIdx.x;
    const float4* __restrict__ p4 = (const float4*)pred;
    const float4* __restrict__ g4 = (const float4*)gt;
    const long long base4 = (long long)b * D4_;

    float pl = 0.f, nl = 0.f, np = 0.f;
    int j = blockIdx.x * 256 + tid;
#pragma unroll
    for (int it = 0; it < ITERS_; ++it, j += BPB_ * 256) {
        float4 xv = p4[base4 + j];
        float4 gv = g4[base4 + j];
        const float xs[4] = {xv.x, xv.y, xv.z, xv.w};
        const float gs[4] = {gv.x, gv.y, gv.z, gv.w};
#pragma unroll
        for (int c = 0; c < 4; ++c) {
            float x = xs[c], g = gs[c];
            float p   = fast_sigmoid(x);
            float omp = 1.0f - p;
            float a   = 1.0f - g;
            float a2  = a * a;               // neg_w = a2*a2
            bool  pos = (g == 1.0f);
            float lp  = __logf(p + EPS_);
            float ln  = __logf(omp + EPS_);
            pl += pos ? lp * omp * omp : 0.0f;
            nl += pos ? 0.0f : ln * p * p * (a2 * a2);
            np += pos ? 1.0f : 0.0f;
        }
    }

    __shared__ float s0[256], s1[256], s2[256];
    s0[tid] = pl; s1[tid] = nl; s2[tid] = np;
    __syncthreads();

    if (tid < 32) {   // wave 0, all 32 lanes active (EXEC all-ones for WMMA)
        float rpl = wmma_reduce256(s0, tid);
        float rnl = wmma_reduce256(s1, tid);
        float rnp = wmma_reduce256(s2, tid);
        if (tid == 0) {
            float* o = partials + ((long long)b * BPB_ + blockIdx.x) * 3;
            o[0] = rpl; o[1] = rnl; o[2] = rnp;
        }
    }
}

// -------------------------------------------------------------------
// Kernel 2: one block per batch. Finalize hm focal loss from partials
// (WMMA tree again), plus the tiny hw-L1 and bu-tag losses
// (gather-then-sigmoid: avoids sigmoiding the full 109 MB bu tensor).
// out layout: [0..15]=hm, [16..31]=hw, [32..47]=bu.
// -------------------------------------------------------------------
__global__ __launch_bounds__(256)
void finalize_losses(const float* __restrict__ partials,
                     const float* __restrict__ hw_pred,
                     const float* __restrict__ hw_gt,
                     const float* __restrict__ bu_pred,
                     const float* __restrict__ bu_gt,
                     float* __restrict__ out) {
    const int b   = blockIdx.x;
    const int tid = threadIdx.x;

    __shared__ float s[256];
    __shared__ float res[3];

    // ---- hm focal finalize: sum BPB_ partial triples ----
    for (int c = 0; c < 3; ++c) {
        s[tid] = (tid < BPB_) ? partials[((long long)b * BPB_ + tid) * 3 + c] : 0.0f;
        __syncthreads();
        if (tid < 32) {
            float r = wmma_reduce256(s, tid);
            if (tid == 0) res[c] = r;
        }
        __syncthreads();
    }
    if (tid == 0) {
        float pl = res[0], nl = res[1], np = res[2];
        out[b] = (np > 0.0f) ? -(pl + nl) / fmaxf(np, 1.0f) : -nl;   // HM_FACTOR = 1
    }

    // ---- hw reg-L1 loss: P_*K_ = 40 gathers ----
    __shared__ float l1s[P_ * K_];
    __shared__ int   wfs[P_ * K_];
    if (tid < P_ * K_) {
        const float* g3 = hw_gt + ((long long)b * P_ * K_ + tid) * 3;
        float val = g3[0];
        int   idx = (int)g3[1];
        bool  wb  = g3[2] > 0.0f;
        float gv  = hw_pred[(long long)b * DHW_ + idx];
        l1s[tid] = wb ? fabsf(gv - val) : 0.0f;
        wfs[tid] = wb ? 1 : 0;
    }
    __syncthreads();
    if (tid == 0) {
        float l1 = 0.0f, npeople = 0.0f;
        for (int p = 0; p < P_; ++p) {
            l1 += l1s[2 * p] + l1s[2 * p + 1];
            npeople += (wfs[2 * p] | wfs[2 * p + 1]) ? 1.0f : 0.0f;
        }
        out[B_ + b] = (npeople > 0.0f) ? (l1 / fmaxf(npeople, 1.0f)) * 0.1f : 0.0f;
    }

    // ---- bu emo-focal tag loss: P_*E_ = 520 gathers, sigmoid only those ----
    __shared__ float bpl[P_ * E_], bnl[P_ * E_], bpos[P_ * E_];
    __shared__ int   bm[P_ * E_];
    for (int t = tid; t < P_ * E_; t += 256) {
        const float* g3 = bu_gt + ((long long)b * P_ * E_ + t) * 3;
        float val = g3[0];
        int   idx = (int)g3[1];
        bool  m   = g3[2] > 0.0f;
        float gsig = fast_sigmoid(bu_pred[(long long)b * D_ + idx]);
        float omg  = 1.0f - gsig;
        bool  pos  = (val == 1.0f) && m;
        bool  neg  = (val <  1.0f) && m;
        bpl[t]  = pos ? __logf(gsig + EPS_) * omg * omg : 0.0f;
        bnl[t]  = neg ? __logf(omg + EPS_) * gsig * gsig : 0.0f;
        bpos[t] = pos ? 1.0f : 0.0f;
        bm[t]   = m ? 1 : 0;
    }
    __syncthreads();
    __shared__ float pper[P_], ppv[P_];
    if (tid < P_) {
        float plp = 0.f, nlp = 0.f, npp = 0.f;
        int anym = 0;
        for (int e = 0; e < E_; ++e) {
            int i = tid * E_ + e;
            plp += bpl[i]; nlp += bnl[i]; npp += bpos[i]; anym |= bm[i];
        }
        float per = (npp == 0.0f) ? -nlp : -(plp + nlp);
        pper[tid] = anym ? per : 0.0f;
        ppv[tid]  = anym ? 1.0f : 0.0f;
    }
    __syncthreads();
    if (tid == 0) {
        float sp = 0.f, sv = 0.f;
        for (int p = 0; p < P_; ++p) { sp += pper[p]; sv += ppv[p]; }
        out[2 * B_ + b] = sp / fmaxf(sv, 1.0f);   // BU_FACTOR = 1
    }
}

extern "C" void kernel_launch(void* const* d_in, const int* in_sizes, int n_in,
                              void* d_out, int out_size, void* d_ws, size_t ws_size,
                              hipStream_t stream) {
    const float* hm_pred = (const float*)d_in[0];
    const float* hm_gt   = (const float*)d_in[1];
    const float* hw_pred = (const float*)d_in[2];
    const float* hw_gt   = (const float*)d_in[3];
    const float* bu_pred = (const float*)d_in[4];
    const float* bu_gt   = (const float*)d_in[5];
    float* out      = (float*)d_out;
    float* partials = (float*)d_ws;   // B_*BPB_*3 floats ~ 40 KB

    dim3 grid1(BPB_, B_);
    hipLaunchKernelGGL(hm_focal_partial, grid1, dim3(256), 0, stream,
                       hm_pred, hm_gt, partials);
    hipLaunchKernelGGL(finalize_losses, dim3(B_), dim3(256), 0, stream,
                       partials, hw_pred, hw_gt, bu_pred, bu_gt, out);
}